// DSimilarity_73684458930900
// MI455X (gfx1250) — compile-verified
//
#include <hip/hip_runtime.h>
#include <math.h>

typedef __attribute__((ext_vector_type(2))) float v2f;
typedef __attribute__((ext_vector_type(4))) float v4f;
typedef __attribute__((ext_vector_type(8))) float v8f;

// Fragment-swizzled storage for a [K x D] operand consumed by WMMA
// (A: D = #rows, indexed by row; B: D = #cols, indexed by col):
//   element (k, idx) lives at  ((k/4)*D + idx)*4 + (k&3)
// so one lane's fragment pair is a single aligned b64, and a wave's whole
// 16x4 fragment is 256B of consecutive memory.

// ---------------------------------------------------------------- utilities
__global__ void zero_f32_kernel(float* __restrict__ p, long n) {
  long i = (long)blockIdx.x * blockDim.x + threadIdx.x;
  long stride = (long)gridDim.x * blockDim.x;
  for (; i < n; i += stride) p[i] = 0.0f;
}

// Scatter unit vectors into fragment-swizzled M1 (A of GEMM2, MP rows) and
// M2t (B of GEMM1, NP cols). Pair p owns k-index p in both -> no atomics.
__global__ void scatter_kernel(const float* __restrict__ u1,
                               const float* __restrict__ u2,
                               const int* __restrict__ i1,
                               const int* __restrict__ i2,
                               float* __restrict__ M1, float* __restrict__ M2t,
                               int npairs, int MP, int NP) {
  int p = blockIdx.x * blockDim.x + threadIdx.x;
  if (p >= npairs) return;
  int r1 = 3 * i1[p];
  int c2 = 3 * i2[p];
  long g4 = (long)(p >> 2) * 4;
  int  kr = p & 3;
#pragma unroll
  for (int a = 0; a < 3; ++a) {
    M1 [(g4 * MP + (long)(r1 + a) * 4) + kr] = u1[p * 3 + a];
    M2t[(g4 * NP + (long)(c2 + a) * 4) + kr] = u2[p * 3 + a];
  }
}

// GG (A of GEMM1): thread (i, g) computes k-group g (j = 4g..4g+3) and writes
// one float4 at ((g*K + i)*4) -> consecutive i threads write a contiguous run.
__global__ void gg_kernel(const float* __restrict__ d1,
                          const float* __restrict__ d2,
                          const float* __restrict__ ls,
                          float* __restrict__ GG, int n) {
  int g = blockIdx.x;                                  // k-group
  int i = blockIdx.y * blockDim.x + threadIdx.x;       // row
  if (i >= n) return;
  float l = ls[0];
  float inv_l2 = 1.0f / (l * l);
  float a = d1[i];
  v4f v;
#pragma unroll
  for (int r = 0; r < 4; ++r) {
    float diff = a - d2[4 * g + r];
    float t = diff * diff * inv_l2;
    v[r] = inv_l2 * (1.0f - t) * expf(-0.5f * t);
  }
  *(v4f*)(GG + ((long)g * n + i) * 4) = v;
}

// ------------------------------------------------- WMMA fp32 GEMM (wave32)
// C[M,N] = A[M,K]*B[K,N]; one wave per 32(M) x 64(N) tile; K stepped in
// groups of 4 with V_WMMA_F32_16X16X4_F32. A/B pre-swizzled (see above):
// each lane's fragment is one b64. Software-pipelined: group g+1's 6 b64
// loads are issued before group g's 8 wmmas.
#define WMMA4(a0, a1, b0, b1, b2, b3)                                          \
  acc00 = __builtin_amdgcn_wmma_f32_16x16x4_f32(false, a0, false, b0, (short)0, acc00, false, false); \
  acc01 = __builtin_amdgcn_wmma_f32_16x16x4_f32(false, a0, false, b1, (short)0, acc01, false, false); \
  acc02 = __builtin_amdgcn_wmma_f32_16x16x4_f32(false, a0, false, b2, (short)0, acc02, false, false); \
  acc03 = __builtin_amdgcn_wmma_f32_16x16x4_f32(false, a0, false, b3, (short)0, acc03, false, false); \
  acc10 = __builtin_amdgcn_wmma_f32_16x16x4_f32(false, a1, false, b0, (short)0, acc10, false, false); \
  acc11 = __builtin_amdgcn_wmma_f32_16x16x4_f32(false, a1, false, b1, (short)0, acc11, false, false); \
  acc12 = __builtin_amdgcn_wmma_f32_16x16x4_f32(false, a1, false, b2, (short)0, acc12, false, false); \
  acc13 = __builtin_amdgcn_wmma_f32_16x16x4_f32(false, a1, false, b3, (short)0, acc13, false, false)

__global__ void __launch_bounds__(32)
wmma_gemm_f32_kernel(const float* __restrict__ A, const float* __restrict__ B,
                     float* __restrict__ C,
                     int aRows, int bCols, int G,
                     int storeSwz, int ldc, int maxRow, int maxCol) {
  const int lane  = threadIdx.x;          // wave32
  const int m     = lane & 15;
  const int half  = lane >> 4;            // 0/1 -> k sub-pair
  const int tileM = blockIdx.y * 32;
  const int tileN = blockIdx.x * 64;

  v8f acc00 = {}, acc01 = {}, acc02 = {}, acc03 = {};
  v8f acc10 = {}, acc11 = {}, acc12 = {}, acc13 = {};

  const float* Ab = A + (long)(tileM + m) * 4 + 2 * half;  // frag row base
  const float* Bb = B + (long)(tileN + m) * 4 + 2 * half;  // frag col base
  const long sA = (long)aRows * 4;                          // floats per group
  const long sB = (long)bCols * 4;

  v2f a0 = *(const v2f*)(Ab);
  v2f a1 = *(const v2f*)(Ab + 64);        // +16 rows
  v2f b0 = *(const v2f*)(Bb);
  v2f b1 = *(const v2f*)(Bb + 64);        // +16 cols
  v2f b2 = *(const v2f*)(Bb + 128);
  v2f b3 = *(const v2f*)(Bb + 192);

  for (int g = 1; g < G; ++g) {
    const float* An = Ab + (long)g * sA;
    const float* Bn = Bb + (long)g * sB;
    v2f na0 = *(const v2f*)(An);
    v2f na1 = *(const v2f*)(An + 64);
    v2f nb0 = *(const v2f*)(Bn);
    v2f nb1 = *(const v2f*)(Bn + 64);
    v2f nb2 = *(const v2f*)(Bn + 128);
    v2f nb3 = *(const v2f*)(Bn + 192);
    WMMA4(a0, a1, b0, b1, b2, b3);
    a0 = na0; a1 = na1;
    b0 = nb0; b1 = nb1; b2 = nb2; b3 = nb3;
  }
  WMMA4(a0, a1, b0, b1, b2, b3);

  // C layout per v8f: VGPR r -> row (r + 8*half), lanes' N = m.
#pragma unroll
  for (int r = 0; r < 8; ++r) {
    int row0 = tileM + r + half * 8;       // acc0x rows
    int row1 = row0 + 16;                  // acc1x rows
    int col  = tileN + m;
    if (storeSwz) {                        // full-tile store into B-swizzled T
      long o0 = ((long)(row0 >> 2) * bCols) * 4 + (row0 & 3);
      long o1 = ((long)(row1 >> 2) * bCols) * 4 + (row1 & 3);
      C[o0 + (long)(col)      * 4] = acc00[r];
      C[o0 + (long)(col + 16) * 4] = acc01[r];
      C[o0 + (long)(col + 32) * 4] = acc02[r];
      C[o0 + (long)(col + 48) * 4] = acc03[r];
      C[o1 + (long)(col)      * 4] = acc10[r];
      C[o1 + (long)(col + 16) * 4] = acc11[r];
      C[o1 + (long)(col + 32) * 4] = acc12[r];
      C[o1 + (long)(col + 48) * 4] = acc13[r];
    } else {                               // bounds-checked row-major store
      if (row0 < maxRow) {
        long base = (long)row0 * ldc;
        if (col      < maxCol) C[base + col]      = acc00[r];
        if (col + 16 < maxCol) C[base + col + 16] = acc01[r];
        if (col + 32 < maxCol) C[base + col + 32] = acc02[r];
        if (col + 48 < maxCol) C[base + col + 48] = acc03[r];
      }
      if (row1 < maxRow) {
        long base = (long)row1 * ldc;
        if (col      < maxCol) C[base + col]      = acc10[r];
        if (col + 16 < maxCol) C[base + col + 16] = acc11[r];
        if (col + 32 < maxCol) C[base + col + 32] = acc12[r];
        if (col + 48 < maxCol) C[base + col + 48] = acc13[r];
      }
    }
  }
}

// ------------------------------------------------------------------- launch
extern "C" void kernel_launch(void* const* d_in, const int* in_sizes, int n_in,
                              void* d_out, int out_size, void* d_ws, size_t ws_size,
                              hipStream_t stream) {
  const float* d1 = (const float*)d_in[0];
  const float* u1 = (const float*)d_in[1];
  const float* d2 = (const float*)d_in[2];
  const float* u2 = (const float*)d_in[3];
  const float* ls = (const float*)d_in[4];
  const int*   i1 = (const int*)d_in[5];
  const int*   i2 = (const int*)d_in[6];

  const int K  = in_sizes[0];                           // n_pairs (4000), %16==0
  const int n3 = (int)llround(sqrt((double)out_size));  // 3*natoms (1500)
  const int MP = (n3 + 31) & ~31;                       // 1504 (=32*47)
  const int NP = (n3 + 63) & ~63;                       // 1536
  const int G  = K / 4;                                 // k-groups

  float* ws  = (float*)d_ws;
  float* GG  = ws;                                      // [K x K ] A-swz  64 MB
  float* M2t = GG  + (long)K * K;                       // [K x NP] B-swz  24 MB
  float* M1  = M2t + (long)K * NP;                      // [MP x K] A-swz  24 MB
  float* T   = M1  + (long)MP * K;                      // [K x NP] B-swz  24 MB
  // ~137 MB total: entire working set is resident in the 192 MB L2.

  // 1) zero M2t + M1 (contiguous)
  long nz = (long)K * NP + (long)MP * K;
  zero_f32_kernel<<<4096, 256, 0, stream>>>(M2t, nz);

  // 2) scatter unit vectors (swizzled)
  scatter_kernel<<<(K + 255) / 256, 256, 0, stream>>>(u1, u2, i1, i2,
                                                      M1, M2t, K, MP, NP);

  // 3) pairwise RBF cross second derivative, A-swizzled, coalesced float4
  gg_kernel<<<dim3(G, (K + 255) / 256), 256, 0, stream>>>(d1, d2, ls, GG, K);

  // 4) T = GG * M2t   (store B-swizzled for GEMM2)
  wmma_gemm_f32_kernel<<<dim3(NP / 64, K / 32), 32, 0, stream>>>(
      GG, M2t, T, K, NP, G, /*swz*/1, 0, 0, 0);

  // 5) out = M1 * T   (plain row-major, bounds-checked)
  wmma_gemm_f32_kernel<<<dim3(NP / 64, MP / 32), 32, 0, stream>>>(
      M1, T, (float*)d_out, MP, NP, G, /*swz*/0, n3, n3, n3);
}